// MultiBoxLoss_69509750719010
// MI455X (gfx1250) — compile-verified
//
#include <hip/hip_runtime.h>
#include <hip/hip_bf16.h>

constexpr int B = 64;
constexpr int P = 32768;
constexpr int C = 8;
constexpr int O = 16;

typedef __attribute__((ext_vector_type(2))) float v2f;
typedef __attribute__((ext_vector_type(8))) float v8f;
typedef __attribute__((ext_vector_type(4))) int   v4i;

#if __has_builtin(__builtin_amdgcn_wmma_f32_16x16x4_f32)
#define HAVE_WMMA_X4 1
#endif

#if __has_builtin(__builtin_amdgcn_global_load_async_to_lds_b128) && \
    __has_builtin(__builtin_amdgcn_s_wait_asynccnt)
#define USE_ASYNC 1
typedef __attribute__((address_space(1))) v4i* gptr_v4i;   // global int4*
typedef __attribute__((address_space(3))) v4i* lptr_v4i;   // LDS int4*
#endif

#if __has_builtin(__builtin_amdgcn_ballot_w32)
#define BALLOT32(x) __builtin_amdgcn_ballot_w32(x)
#else
#define BALLOT32(x) ((unsigned)__ballot(x))
#endif

// ---------------------------------------------------------------------------
// Kernel 1: per-batch matching (IoU, argmaxes, scatter), conf_t, smooth-L1 loc
// loss, num_pos.  One block (1024 thr) per batch.
// ---------------------------------------------------------------------------
__global__ void k_match(const float* __restrict__ loc,
                        const float* __restrict__ priors,
                        const float* __restrict__ targets,
                        float* __restrict__ ov, int* __restrict__ ti,
                        int* __restrict__ ct, int* __restrict__ np,
                        float* __restrict__ llocB)
{
    const int b = blockIdx.x, tid = threadIdx.x;
    const int lane = tid & 31, wid = tid >> 5;
    __shared__ float s_t0[O], s_t1[O], s_lab[O];
    __shared__ unsigned long long s_key[O];
    __shared__ float s_f[32];
    __shared__ int   s_i[32];

    if (tid < O) {
        s_t0[tid]  = targets[(b * O + tid) * 3 + 0];
        s_t1[tid]  = targets[(b * O + tid) * 3 + 1];
        s_lab[tid] = targets[(b * O + tid) * 3 + 2];
        s_key[tid] = 0ull;
    }
    __syncthreads();

    unsigned long long key[O];
#pragma unroll
    for (int o = 0; o < O; ++o) key[o] = 0ull;

    const long base = (long)b * P;
    for (int p = tid; p < P; p += 1024) {
        float pc = priors[2 * p], pw = priors[2 * p + 1];
        float pmin = pc - 0.5f * pw, pmax = pc + 0.5f * pw;
        float bo = -1.0f; int bt = 0;
#pragma unroll
        for (int o = 0; o < O; ++o) {
            float t0 = s_t0[o], t1 = s_t1[o];
            float inter = fminf(t1, pmax) - fmaxf(t0, pmin);
            inter = fmaxf(inter, 0.0f);
            float uni = (t1 - t0) + pw - inter;
            float iou = inter / uni;
            if (iou > bo) { bo = iou; bt = o; }   // first-max over truths
            unsigned long long kk =
                ((unsigned long long)__float_as_uint(iou) << 32) |
                (unsigned long long)(0xFFFFFFFFu - (unsigned)p);  // tie -> low p
            if (kk > key[o]) key[o] = kk;
        }
        ov[base + p] = bo;
        ti[base + p] = bt;
    }
    // reduce per-truth best-prior keys: wave shfl then LDS atomicMax
#pragma unroll
    for (int o = 0; o < O; ++o) {
        unsigned long long kk = key[o];
        for (int off = 16; off; off >>= 1) {
            unsigned long long t = __shfl_down(kk, off, 32);
            if (t > kk) kk = t;
        }
        if (lane == 0) atomicMax(&s_key[o], kk);
    }
    __syncthreads();
    if (tid == 0) {   // sequential scatter (last truth wins on duplicates)
        for (int o = 0; o < O; ++o) {
            unsigned p = 0xFFFFFFFFu - (unsigned)(s_key[o] & 0xFFFFFFFFull);
            ov[base + p] = 2.0f;
            ti[base + p] = o;
        }
    }
    __syncthreads();

    float lsum = 0.f; int pcnt = 0;
    for (int p = tid; p < P; p += 1024) {
        float o2 = ov[base + p];
        int t = ti[base + p];
        int c = (o2 < 0.5f) ? 0 : ((int)s_lab[t] + 1);
        ct[base + p] = c;
        if (c > 0) {
            ++pcnt;
            float pcn = priors[2 * p], pwd = priors[2 * p + 1];
            float t0 = s_t0[t], t1 = s_t1[t];
            float gc = ((t0 + t1) * 0.5f - pcn) / (0.1f * pwd);
            float gw = logf((t1 - t0) / pwd) / 0.2f;
            float d0 = loc[(base + p) * 2 + 0] - gc;
            float d1 = loc[(base + p) * 2 + 1] - gw;
            float a0 = fabsf(d0), a1 = fabsf(d1);
            lsum += (a0 < 1.f ? 0.5f * d0 * d0 : a0 - 0.5f) +
                    (a1 < 1.f ? 0.5f * d1 * d1 : a1 - 0.5f);
        }
    }
    for (int off = 16; off; off >>= 1) {
        lsum += __shfl_down(lsum, off, 32);
        pcnt += __shfl_down(pcnt, off, 32);
    }
    if (lane == 0) { s_f[wid] = lsum; s_i[wid] = pcnt; }
    __syncthreads();
    if (tid == 0) {
        float L = 0.f; int N = 0;
        for (int w = 0; w < 32; ++w) { L += s_f[w]; N += s_i[w]; }
        np[b] = N;
        llocB[b] = L;
    }
}

// ---------------------------------------------------------------------------
// Kernel 2: logsumexp + CE per prior.  One wave handles a 16-prior tile; the
// 8-class exp-sum is done with two chained V_WMMA_F32_16X16X4_F32 (B = ones).
// A-matrix layout (ISA 7.12.2, 32-bit A 16x4): lanes0-15 hold K0,K1; lanes
// 16-31 hold K2,K3, M = lane&15.  D layout: VGPR r = row r (lanes0-15) /
// row 8+r (lanes16-31); columns identical since B is all-ones.
// ---------------------------------------------------------------------------
__global__ void k_lse(const float* __restrict__ conf,
                      const int* __restrict__ ct,
                      float* __restrict__ ce, float* __restrict__ lc)
{
    const int tid = threadIdx.x;
    const int wv = tid >> 5, L = tid & 31;
    const long tile = (long)blockIdx.x * 8 + wv;   // B*P/16 tiles total
    const int L16 = L & 15, hi = L >> 4;
    const long row = tile * 16 + L16;              // flattened b*P + p
    const float* cp = conf + row * 8;

    float a0 = cp[hi * 2 + 0], a1 = cp[hi * 2 + 1];         // classes 0..3
    float b0 = cp[4 + hi * 2 + 0], b1 = cp[4 + hi * 2 + 1]; // classes 4..7
    float m = fmaxf(fmaxf(a0, a1), fmaxf(b0, b1));
    m = fmaxf(m, __shfl_xor(m, 16, 32));          // row max over 8 classes
    float e0 = expf(a0 - m), e1 = expf(a1 - m);
    float e2 = expf(b0 - m), e3 = expf(b1 - m);

    float S;
#if defined(HAVE_WMMA_X4)
    v2f A1; A1[0] = e0; A1[1] = e1;
    v2f A2; A2[0] = e2; A2[1] = e3;
    v2f Bv; Bv[0] = 1.0f; Bv[1] = 1.0f;
    v8f accv = {};
    accv = __builtin_amdgcn_wmma_f32_16x16x4_f32(false, A1, false, Bv,
                                                 (short)0, accv, false, false);
    accv = __builtin_amdgcn_wmma_f32_16x16x4_f32(false, A2, false, Bv,
                                                 (short)0, accv, false, false);
    int e = L & 7;
    float x = accv[e];                 // rowsum(e) (lo half) / rowsum(8+e) (hi)
    float y = __shfl_xor(x, 16, 32);   // the other half's row group
    bool usex = (((L >> 3) & 1) == ((L >> 4) & 1));
    S = usex ? x : y;                  // S = rowsum(L & 15)
#else
    S = e0 + e1 + e2 + e3;
    S += __shfl_xor(S, 16, 32);
#endif

    float lse = m + logf(S);
    int c = ct[row];
    float g = cp[c];
    float cev = lse - g;
    if (L < 16) {
        ce[row] = cev;
        lc[row] = (c > 0) ? 0.0f : cev;
    }
}

// ---------------------------------------------------------------------------
// Kernel 3: per-batch radix-select of the num_neg-th largest loss_c.  Batch's
// 128KB of loss_c is staged into LDS (async global->LDS on CDNA5), then 32
// bit-test counting passes run entirely out of LDS.  loss_c >= 0, so uint
// bit-pattern order == float order.
// ---------------------------------------------------------------------------
__global__ void k_select(const float* __restrict__ lc,
                         const int* __restrict__ np,
                         unsigned* __restrict__ Tq,
                         int* __restrict__ quota)
{
    const int b = blockIdx.x, tid = threadIdx.x;
    const int lane = tid & 31;
    __shared__ float s_loss[P];        // 128 KB (gfx1250 LDS is large)
    __shared__ int s_tot;
    const float* src = lc + (long)b * P;

#if defined(USE_ASYNC)
    {
        gptr_v4i g = (gptr_v4i)src;             // global int4*, const dropped
        lptr_v4i l = (lptr_v4i)&s_loss[0];      // LDS int4*
        for (int i = tid; i < P / 4; i += 1024)
            __builtin_amdgcn_global_load_async_to_lds_b128(g + i, l + i, 0, 0);
        __builtin_amdgcn_s_wait_asynccnt(0);
    }
#else
    for (int i = tid; i < P; i += 1024) s_loss[i] = src[i];
#endif
    __syncthreads();

    int k = np[b] * 3;
    if (k > P - 1) k = P - 1;

    auto countGE = [&](unsigned t) -> int {
        __syncthreads();
        if (tid == 0) s_tot = 0;
        __syncthreads();
        int c = 0;
        for (int i = tid; i < P; i += 1024)
            c += (__float_as_uint(s_loss[i]) >= t) ? 1 : 0;
        for (int off = 16; off; off >>= 1) c += __shfl_down(c, off, 32);
        if (lane == 0) atomicAdd(&s_tot, c);
        __syncthreads();
        return s_tot;
    };

    unsigned prefix = 0;
    int G = 0;
    if (k > 0) {
        for (int bit = 31; bit >= 0; --bit) {
            unsigned t = prefix | (1u << bit);
            if (countGE(t) >= k) prefix = t;   // k-th largest bit pattern
        }
        G = countGE(prefix + 1);               // strictly greater than T
    } else {
        prefix = 0xFFFFFFFFu;
    }
    if (tid == 0) {
        Tq[b] = prefix;
        quota[b] = (k > 0) ? (k - G) : 0;      // equals selected in index order
    }
}

// ---------------------------------------------------------------------------
// Kernel 4: selection (pos | rank<num_neg with stable ties) + CE sum/batch.
// Index-ordered quota among equals via ballot prefix scan.
// ---------------------------------------------------------------------------
__global__ void k_mine(const float* __restrict__ lc,
                       const float* __restrict__ ce,
                       const int* __restrict__ ct,
                       const unsigned* __restrict__ Tq,
                       const int* __restrict__ quota,
                       float* __restrict__ lconfB)
{
    const int b = blockIdx.x, tid = threadIdx.x;
    const int lane = tid & 31, wid = tid >> 5;
    __shared__ int s_wc[32];
    __shared__ int s_base;
    __shared__ float s_f[32];
    const long base = (long)b * P;
    const unsigned T = Tq[b];
    const int Q = quota[b];
    if (tid == 0) s_base = 0;
    __syncthreads();

    float sum = 0.f;
    for (int s0 = 0; s0 < P; s0 += 1024) {
        int p = s0 + tid;
        unsigned u = __float_as_uint(lc[base + p]);
        bool eq = (u == T), gt = (u > T);
        unsigned mask = BALLOT32(eq);
        int preLane = __popc(mask & ((1u << lane) - 1u));
        if (lane == 0) s_wc[wid] = __popc(mask);
        __syncthreads();
        int pre = s_base + preLane;
        for (int w = 0; w < wid; ++w) pre += s_wc[w];
        bool selEq = eq && (pre < Q);
        bool pos = ct[base + p] > 0;
        if (pos | gt | selEq) sum += ce[base + p];
        __syncthreads();
        if (tid == 0) {
            int t = 0;
            for (int w = 0; w < 32; ++w) t += s_wc[w];
            s_base += t;
        }
        __syncthreads();
    }
    for (int off = 16; off; off >>= 1) sum += __shfl_down(sum, off, 32);
    if (lane == 0) s_f[wid] = sum;
    __syncthreads();
    if (tid == 0) {
        float L = 0.f;
        for (int w = 0; w < 32; ++w) L += s_f[w];
        lconfB[b] = L;
    }
}

// ---------------------------------------------------------------------------
// Kernel 5: deterministic fixed-order finalize.
// ---------------------------------------------------------------------------
__global__ void k_final(const int* __restrict__ np,
                        const float* __restrict__ llocB,
                        const float* __restrict__ lconfB,
                        float* __restrict__ out)
{
    if (threadIdx.x == 0 && blockIdx.x == 0) {
        float N = 0.f, LL = 0.f, LC = 0.f;
        for (int b = 0; b < B; ++b) {
            N += (float)np[b];
            LL += llocB[b];
            LC += lconfB[b];
        }
        out[0] = LL / N;
        out[1] = LC / N;
    }
}

extern "C" void kernel_launch(void* const* d_in, const int* in_sizes, int n_in,
                              void* d_out, int out_size, void* d_ws, size_t ws_size,
                              hipStream_t stream)
{
    const float* loc     = (const float*)d_in[0];   // (B,P,2)
    const float* conf    = (const float*)d_in[1];   // (B,P,C)
    const float* priors  = (const float*)d_in[2];   // (P,2)
    const float* targets = (const float*)d_in[3];   // (B,O,3)
    float* out = (float*)d_out;                     // 2 floats

    const size_t BP = (size_t)B * P;
    char* W = (char*)d_ws;
    float*    ws_ov = (float*)(W + BP * 0);         // best_truth_overlap
    int*      ws_ti = (int*)  (W + BP * 4);         // best_truth_idx
    int*      ws_ct = (int*)  (W + BP * 8);         // conf_t
    float*    ws_ce = (float*)(W + BP * 12);        // cross-entropy per prior
    float*    ws_lc = (float*)(W + BP * 16);        // mining loss_c
    char* S = W + BP * 20;
    int*      ws_np  = (int*)     (S + 0);          // [B]
    unsigned* ws_T   = (unsigned*)(S + 256);        // [B]
    int*      ws_q   = (int*)     (S + 512);        // [B]
    float*    ws_llb = (float*)   (S + 768);        // [B] loc-loss partials
    float*    ws_lcb = (float*)   (S + 1024);       // [B] conf-loss partials

    k_match <<<B, 1024, 0, stream>>>(loc, priors, targets, ws_ov, ws_ti,
                                     ws_ct, ws_np, ws_llb);
    k_lse   <<<(B * P / 16) / 8, 256, 0, stream>>>(conf, ws_ct, ws_ce, ws_lc);
    k_select<<<B, 1024, 0, stream>>>(ws_lc, ws_np, ws_T, ws_q);
    k_mine  <<<B, 1024, 0, stream>>>(ws_lc, ws_ce, ws_ct, ws_T, ws_q, ws_lcb);
    k_final <<<1, 64, 0, stream>>>(ws_np, ws_llb, ws_lcb, out);
}